// HierarchicalDiffPoolGNN_72937134621226
// MI455X (gfx1250) — compile-verified
//
#include <hip/hip_runtime.h>
#include <hip/hip_bf16.h>

// ---------------------------------------------------------------------------
// HierarchicalDiffPoolGNN for MI455X (gfx1250, wave32, WMMA).
// All dense algebra runs on V_WMMA_F32_16X16X4_F32 (fp32 in/out, matches
// reference precision). One wave per 16x64 C-tile: 4 independent WMMA
// accumulator chains (hides XDL latency) + 4x A-operand reuse.
// ---------------------------------------------------------------------------

typedef float v2f __attribute__((ext_vector_type(2)));
typedef float v8f __attribute__((ext_vector_type(8)));

__device__ __forceinline__ v8f wmma4(v2f a, v2f b, v8f c) {
  // 8 args: (neg_a, A, neg_b, B, c_mod, C, reuse_a, reuse_b)
  return __builtin_amdgcn_wmma_f32_16x16x4_f32(false, a, false, b, (short)0, c,
                                               false, false);
}

constexpr int B_  = 8;
constexpr int N_  = 2048;
constexpr int E_  = 16384;
constexpr int DIN = 128;
constexpr int F_  = 256;   // H == Dout == C == 256
constexpr int C_  = 256;

// d_out layout (floats), reference tuple order
constexpr size_t OUT_PF   = 0;                                   // B*C*Dout
constexpr size_t OUT_PEI  = OUT_PF  + (size_t)B_*C_*F_;          // B*2*C*C
constexpr size_t OUT_PNM  = OUT_PEI + (size_t)B_*2*C_*C_;        // B*C
constexpr size_t OUT_PEM  = OUT_PNM + (size_t)B_*C_;             // B*C*C
constexpr size_t OUT_LINK = OUT_PEM + (size_t)B_*C_*C_;
constexpr size_t OUT_ENT  = OUT_LINK + 1;
constexpr size_t OUT_ORTHO= OUT_ENT  + 1;

// ---------------------------------------------------------------------------
// Utility kernels
// ---------------------------------------------------------------------------
__global__ void fill_f32(float* p, float v, size_t n) {
  size_t i = (size_t)blockIdx.x * blockDim.x + threadIdx.x;
  size_t st = (size_t)gridDim.x * blockDim.x;
  for (; i < n; i += st) p[i] = v;
}

__global__ void deg_kernel(const int* __restrict__ ei,
                           const float* __restrict__ em,
                           float* __restrict__ deg) {
  int idx = blockIdx.x * blockDim.x + threadIdx.x;   // b*E + e
  if (idx >= B_ * E_) return;
  int b = idx / E_, e = idx % E_;
  int t = ei[(size_t)b * 2 * E_ + E_ + e];
  atomicAdd(&deg[(size_t)b * N_ + t], em[idx]);
}

__global__ void aggregate_kernel(const float* __restrict__ X,
                                 const int* __restrict__ ei,
                                 const float* __restrict__ em,
                                 float* __restrict__ agg, int K) {
  int chunk = K >> 2;
  size_t idx = (size_t)blockIdx.x * blockDim.x + threadIdx.x;
  size_t tot = (size_t)B_ * E_ * chunk;
  if (idx >= tot) return;
  int d4 = (int)(idx % chunk) * 4;
  size_t be = idx / chunk;
  int b = (int)(be / E_), e = (int)(be % E_);
  int s = ei[(size_t)b * 2 * E_ + e];
  int t = ei[(size_t)b * 2 * E_ + E_ + e];
  float m = em[(size_t)b * E_ + e];
  const float* xp = X + ((size_t)b * N_ + s) * K + d4;
  float* ap = agg + ((size_t)b * N_ + t) * K + d4;
  atomicAdd(ap + 0, xp[0] * m);
  atomicAdd(ap + 1, xp[1] * m);
  atomicAdd(ap + 2, xp[2] * m);
  atomicAdd(ap + 3, xp[3] * m);
}

__global__ void mean_div_kernel(const float* __restrict__ agg,
                                const float* __restrict__ deg,
                                float* __restrict__ meanb, int K) {
  size_t idx = (size_t)blockIdx.x * blockDim.x + threadIdx.x;
  size_t tot = (size_t)B_ * N_ * K;
  if (idx >= tot) return;
  size_t row = idx / K;                       // b*N + n
  meanb[idx] = agg[idx] / fmaxf(deg[row], 1.0f);
}

// ---------------------------------------------------------------------------
// WMMA GEMM kernels. One wave owns a 16x64 output tile -> 4 accumulators.
// VGPR layouts per CDNA5 ISA 7.12.2:
//   A 16x4 : lane l(0..15) VGPR{0,1} = A[r0+l][k, k+1]; lanes 16..31 -> k+2,k+3
//   B 4x16 : VGPR{0,1} lanes 0..15 = B[k,k+1][c0+l];   lanes 16..31 -> k+2,k+3
//   C 16x16: VGPR v = rows r0+v (lanes 0..15) and r0+v+8 (lanes 16..31)
// ---------------------------------------------------------------------------
__global__ __launch_bounds__(32)
void sage_gemm_wmma(const float* __restrict__ X, const float* __restrict__ Mn,
                    const float* __restrict__ Ws, const float* __restrict__ Wn,
                    const float* __restrict__ bias,
                    const float* __restrict__ nmask,
                    float* __restrict__ Out, int K) {
  int r0 = blockIdx.x * 16, c0 = blockIdx.y * 64, b = blockIdx.z;
  int lane = threadIdx.x, half = lane >> 4, l = lane & 15;
  const float* Xb = X + ((size_t)b * N_) * K;
  const float* Mb = Mn + ((size_t)b * N_) * K;
  v8f a0 = {}, a1 = {}, a2 = {}, a3 = {};
  __builtin_prefetch(&Ws[c0 + l], 0, 1);
  __builtin_prefetch(&Wn[c0 + l], 0, 1);
  for (int k = 0; k < K; k += 4) {
    int ka = k + half * 2;
    v2f a, b0, b1, b2, b3;
    a.x = Xb[(size_t)(r0 + l) * K + ka];
    a.y = Xb[(size_t)(r0 + l) * K + ka + 1];
    b0.x = Ws[(size_t)ka * F_ + c0 + l];
    b0.y = Ws[(size_t)(ka + 1) * F_ + c0 + l];
    b1.x = Ws[(size_t)ka * F_ + c0 + 16 + l];
    b1.y = Ws[(size_t)(ka + 1) * F_ + c0 + 16 + l];
    b2.x = Ws[(size_t)ka * F_ + c0 + 32 + l];
    b2.y = Ws[(size_t)(ka + 1) * F_ + c0 + 32 + l];
    b3.x = Ws[(size_t)ka * F_ + c0 + 48 + l];
    b3.y = Ws[(size_t)(ka + 1) * F_ + c0 + 48 + l];
    a0 = wmma4(a, b0, a0);
    a1 = wmma4(a, b1, a1);
    a2 = wmma4(a, b2, a2);
    a3 = wmma4(a, b3, a3);
  }
  for (int k = 0; k < K; k += 4) {
    int ka = k + half * 2;
    v2f a, b0, b1, b2, b3;
    a.x = Mb[(size_t)(r0 + l) * K + ka];
    a.y = Mb[(size_t)(r0 + l) * K + ka + 1];
    b0.x = Wn[(size_t)ka * F_ + c0 + l];
    b0.y = Wn[(size_t)(ka + 1) * F_ + c0 + l];
    b1.x = Wn[(size_t)ka * F_ + c0 + 16 + l];
    b1.y = Wn[(size_t)(ka + 1) * F_ + c0 + 16 + l];
    b2.x = Wn[(size_t)ka * F_ + c0 + 32 + l];
    b2.y = Wn[(size_t)(ka + 1) * F_ + c0 + 32 + l];
    b3.x = Wn[(size_t)ka * F_ + c0 + 48 + l];
    b3.y = Wn[(size_t)(ka + 1) * F_ + c0 + 48 + l];
    a0 = wmma4(a, b0, a0);
    a1 = wmma4(a, b1, a1);
    a2 = wmma4(a, b2, a2);
    a3 = wmma4(a, b3, a3);
  }
  v8f accs[4] = {a0, a1, a2, a3};
#pragma unroll
  for (int j = 0; j < 4; j++) {
#pragma unroll
    for (int v = 0; v < 8; v++) {
      int row = r0 + v + half * 8;
      int col = c0 + 16 * j + l;
      float val = accs[j][v] + bias[col];
      val = fmaxf(val, 0.0f) * nmask[(size_t)b * N_ + row];
      Out[((size_t)b * N_ + row) * F_ + col] = val;
    }
  }
}

// Out[M,F] = A^T * B where A stored K x M (lda), B stored K x F (ldb)
__global__ __launch_bounds__(32)
void gemm_tn_wmma(const float* __restrict__ A, const float* __restrict__ Bm,
                  float* __restrict__ Out, int K, int lda, int ldb, int ldo,
                  size_t sA, size_t sB, size_t sO) {
  int r0 = blockIdx.x * 16, c0 = blockIdx.y * 64;
  const float* Ab = A + (size_t)blockIdx.z * sA;
  const float* Bb = Bm + (size_t)blockIdx.z * sB;
  float* Ob = Out + (size_t)blockIdx.z * sO;
  int lane = threadIdx.x, half = lane >> 4, l = lane & 15;
  v8f a0 = {}, a1 = {}, a2 = {}, a3 = {};
  for (int k = 0; k < K; k += 4) {
    int ka = k + half * 2;
    v2f a, b0, b1, b2, b3;
    a.x = Ab[(size_t)ka * lda + r0 + l];
    a.y = Ab[(size_t)(ka + 1) * lda + r0 + l];
    b0.x = Bb[(size_t)ka * ldb + c0 + l];
    b0.y = Bb[(size_t)(ka + 1) * ldb + c0 + l];
    b1.x = Bb[(size_t)ka * ldb + c0 + 16 + l];
    b1.y = Bb[(size_t)(ka + 1) * ldb + c0 + 16 + l];
    b2.x = Bb[(size_t)ka * ldb + c0 + 32 + l];
    b2.y = Bb[(size_t)(ka + 1) * ldb + c0 + 32 + l];
    b3.x = Bb[(size_t)ka * ldb + c0 + 48 + l];
    b3.y = Bb[(size_t)(ka + 1) * ldb + c0 + 48 + l];
    __builtin_prefetch(&Bb[(size_t)(ka + 4) * ldb + c0 + l], 0, 1);
    a0 = wmma4(a, b0, a0);
    a1 = wmma4(a, b1, a1);
    a2 = wmma4(a, b2, a2);
    a3 = wmma4(a, b3, a3);
  }
  v8f accs[4] = {a0, a1, a2, a3};
#pragma unroll
  for (int j = 0; j < 4; j++)
#pragma unroll
    for (int v = 0; v < 8; v++)
      Ob[(size_t)(r0 + v + half * 8) * ldo + c0 + 16 * j + l] = accs[j][v];
}

// Out[M,F] = A * B, A stored M x K (lda), B stored K x F (ldb)
__global__ __launch_bounds__(32)
void gemm_nn_wmma(const float* __restrict__ A, const float* __restrict__ Bm,
                  float* __restrict__ Out, int K, int lda, int ldb, int ldo) {
  int r0 = blockIdx.x * 16, c0 = blockIdx.y * 64;
  int lane = threadIdx.x, half = lane >> 4, l = lane & 15;
  v8f a0 = {}, a1 = {}, a2 = {}, a3 = {};
  for (int k = 0; k < K; k += 4) {
    int ka = k + half * 2;
    v2f a, b0, b1, b2, b3;
    a.x = A[(size_t)(r0 + l) * lda + ka];
    a.y = A[(size_t)(r0 + l) * lda + ka + 1];
    b0.x = Bm[(size_t)ka * ldb + c0 + l];
    b0.y = Bm[(size_t)(ka + 1) * ldb + c0 + l];
    b1.x = Bm[(size_t)ka * ldb + c0 + 16 + l];
    b1.y = Bm[(size_t)(ka + 1) * ldb + c0 + 16 + l];
    b2.x = Bm[(size_t)ka * ldb + c0 + 32 + l];
    b2.y = Bm[(size_t)(ka + 1) * ldb + c0 + 32 + l];
    b3.x = Bm[(size_t)ka * ldb + c0 + 48 + l];
    b3.y = Bm[(size_t)(ka + 1) * ldb + c0 + 48 + l];
    a0 = wmma4(a, b0, a0);
    a1 = wmma4(a, b1, a1);
    a2 = wmma4(a, b2, a2);
    a3 = wmma4(a, b3, a3);
  }
  v8f accs[4] = {a0, a1, a2, a3};
#pragma unroll
  for (int j = 0; j < 4; j++)
#pragma unroll
    for (int v = 0; v < 8; v++)
      Out[(size_t)(r0 + v + half * 8) * ldo + c0 + 16 * j + l] = accs[j][v];
}

// ---------------------------------------------------------------------------
// Softmax + mask + renorm; column sums; adjacency scatter; ortho reduce
// ---------------------------------------------------------------------------
__global__ __launch_bounds__(256)
void softmax_kernel(const float* __restrict__ alog,
                    const float* __restrict__ nmask, float* __restrict__ S) {
  int row = blockIdx.x;  // b*N + n
  int t = threadIdx.x;
  __shared__ float sh[256];
  float v = alog[(size_t)row * C_ + t];
  sh[t] = v; __syncthreads();
  for (int o = 128; o; o >>= 1) { if (t < o) sh[t] = fmaxf(sh[t], sh[t + o]); __syncthreads(); }
  float m = sh[0]; __syncthreads();
  float e = expf(v - m);
  sh[t] = e; __syncthreads();
  for (int o = 128; o; o >>= 1) { if (t < o) sh[t] += sh[t + o]; __syncthreads(); }
  float s = sh[0]; __syncthreads();
  float sm = (e / s) * nmask[row];
  sh[t] = sm; __syncthreads();
  for (int o = 128; o; o >>= 1) { if (t < o) sh[t] += sh[t + o]; __syncthreads(); }
  float s2 = sh[0];
  S[(size_t)row * C_ + t] = sm / fmaxf(s2, 1e-8f);
}

__global__ void colsum_kernel(const float* __restrict__ S,
                              float* __restrict__ colsum,
                              float* __restrict__ out_pnm) {
  int idx = blockIdx.x * blockDim.x + threadIdx.x;  // b*C + c
  if (idx >= B_ * C_) return;
  int b = idx >> 8, c = idx & 255;
  const float* p = S + (size_t)b * N_ * C_ + c;
  float s = 0.0f;
  for (int n = 0; n < N_; n++) s += p[(size_t)n * C_];
  colsum[idx] = s;
  out_pnm[idx] = (s > 1e-6f) ? 1.0f : 0.0f;
}

__global__ void scatter_adj_kernel(const int* __restrict__ ei,
                                   const float* __restrict__ em,
                                   float* __restrict__ A, int b) {
  int e = blockIdx.x * blockDim.x + threadIdx.x;
  if (e >= E_) return;
  int s = ei[(size_t)b * 2 * E_ + e];
  int t = ei[(size_t)b * 2 * E_ + E_ + e];
  float m = em[(size_t)b * E_ + e];
  // non-negative float max == int max on the bit pattern
  atomicMax((int*)&A[(size_t)s * N_ + t], __float_as_int(m));
}

__global__ __launch_bounds__(256)
void reduce_ortho_kernel(const float* __restrict__ gram,
                         float* __restrict__ part_ortho) {
  int b = blockIdx.y;
  int idx = blockIdx.x * 256 + threadIdx.x;  // 0..C*C-1
  float g = gram[(size_t)b * C_ * C_ + idx];
  int r = idx >> 8, c = idx & 255;
  float d = g - ((r == c) ? 1.0f : 0.0f);
  __shared__ float sh[256];
  sh[threadIdx.x] = d * d; __syncthreads();
  for (int o = 128; o; o >>= 1) { if (threadIdx.x < o) sh[threadIdx.x] += sh[threadIdx.x + o]; __syncthreads(); }
  if (threadIdx.x == 0) atomicAdd(&part_ortho[b], sh[0]);
}

// ---------------------------------------------------------------------------
// Stable boolean-sort compaction (argsort of !flat == trues first, in order)
// ---------------------------------------------------------------------------
__global__ __launch_bounds__(1024)
void compact_kernel(const float* __restrict__ padj, float* __restrict__ pei,
                    float* __restrict__ pem) {
  int b = blockIdx.x;
  int t = threadIdx.x;
  const float* P = padj + (size_t)b * C_ * C_;
  __shared__ int sc[1024];
  int base = t * 64;
  int cnt = 0;
  for (int j = 0; j < 64; j++) cnt += (P[base + j] > 1e-6f) ? 1 : 0;
  sc[t] = cnt; __syncthreads();
  for (int off = 1; off < 1024; off <<= 1) {
    int v = (t >= off) ? sc[t - off] : 0;
    __syncthreads();
    sc[t] += v;
    __syncthreads();
  }
  int pos = sc[t] - cnt;  // exclusive prefix of trues
  float* ps = pei + (size_t)b * 2 * C_ * C_;
  float* pt = ps + C_ * C_;
  float* pm = pem + (size_t)b * C_ * C_;
  for (int j = 0; j < 64; j++) {
    int idx = base + j;
    if (P[idx] > 1e-6f) {
      ps[pos] = (float)(idx >> 8);
      pt[pos] = (float)(idx & 255);
      pm[pos] = 1.0f;
      pos++;
    }
  }
}

// ---------------------------------------------------------------------------
// Link prediction: per wave 16 edges; gather 16x512 tile to LDS, WMMA to
// 16x256 hidden (relu) with 4 parallel column chains, dot with W2, sigmoid,
// masked NLL accumulation.
// ---------------------------------------------------------------------------
__global__ __launch_bounds__(32)
void linkpred_kernel(const float* __restrict__ emb, const int* __restrict__ ei,
                     const float* __restrict__ em, const float* __restrict__ W1,
                     const float* __restrict__ b1, const float* __restrict__ W2,
                     const float* __restrict__ b2,
                     float* __restrict__ part_nll) {
  __shared__ float At[16][512];
  __shared__ float Ht[16][256];
  __shared__ float red[32];
  int b = blockIdx.y;
  int e0 = blockIdx.x * 16;
  int lane = threadIdx.x;
  const int* src = ei + (size_t)b * 2 * E_;
  const int* tgt = src + E_;
  const float* embB = emb + (size_t)b * N_ * F_;
  for (int i = lane; i < 16 * 512; i += 32) {
    int r = i >> 9;
    int k = i & 511;
    int node = (k < 256) ? src[e0 + r] : tgt[e0 + r];
    At[r][k] = embB[(size_t)node * F_ + (k & 255)];
  }
  __syncthreads();

  int half = lane >> 4, l = lane & 15;
  for (int c0 = 0; c0 < 256; c0 += 64) {
    v8f a0 = {}, a1 = {}, a2 = {}, a3 = {};
    for (int k = 0; k < 512; k += 4) {
      int ka = k + half * 2;
      v2f a, b0, b1v, b2v, b3v;
      a.x = At[l][ka];
      a.y = At[l][ka + 1];
      b0.x = W1[(size_t)ka * 256 + c0 + l];
      b0.y = W1[(size_t)(ka + 1) * 256 + c0 + l];
      b1v.x = W1[(size_t)ka * 256 + c0 + 16 + l];
      b1v.y = W1[(size_t)(ka + 1) * 256 + c0 + 16 + l];
      b2v.x = W1[(size_t)ka * 256 + c0 + 32 + l];
      b2v.y = W1[(size_t)(ka + 1) * 256 + c0 + 32 + l];
      b3v.x = W1[(size_t)ka * 256 + c0 + 48 + l];
      b3v.y = W1[(size_t)(ka + 1) * 256 + c0 + 48 + l];
      a0 = wmma4(a, b0, a0);
      a1 = wmma4(a, b1v, a1);
      a2 = wmma4(a, b2v, a2);
      a3 = wmma4(a, b3v, a3);
    }
    v8f accs[4] = {a0, a1, a2, a3};
#pragma unroll
    for (int j = 0; j < 4; j++)
#pragma unroll
      for (int v = 0; v < 8; v++) {
        int row = v + half * 8;                   // edge within tile
        int col = c0 + 16 * j + l;
        Ht[row][col] = fmaxf(accs[j][v] + b1[col], 0.0f);
      }
  }
  __syncthreads();

  // p = sigmoid(h . W2 + b2); nll = min(softplus(-d), 100); split K over halves
  float dot = 0.0f;
  for (int c = half * 128; c < half * 128 + 128; c++) dot += Ht[l][c] * W2[c];
  red[lane] = dot; __syncthreads();
  if (half == 0) {
    float d = red[l] + red[l + 16] + b2[0];
    float nl = (d > 0.0f) ? log1pf(expf(-d)) : (-d + log1pf(expf(d)));
    nl = fminf(nl, 100.0f);
    red[l] = nl * em[(size_t)b * E_ + e0 + l];
  }
  __syncthreads();
  if (lane == 0) {
    float s = 0.0f;
    for (int i = 0; i < 16; i++) s += red[i];
    atomicAdd(&part_nll[b], s);
  }
}

__global__ __launch_bounds__(256)
void emsum_kernel(const float* __restrict__ em, float* __restrict__ part_em) {
  int b = blockIdx.x, t = threadIdx.x;
  __shared__ float sh[256];
  float s = 0.0f;
  for (int e = t; e < E_; e += 256) s += em[(size_t)b * E_ + e];
  sh[t] = s; __syncthreads();
  for (int o = 128; o; o >>= 1) { if (t < o) sh[t] += sh[t + o]; __syncthreads(); }
  if (t == 0) part_em[b] = sh[0];
}

__global__ __launch_bounds__(256)
void finalize_kernel(const float* __restrict__ colsum,
                     const float* __restrict__ part_nll,
                     const float* __restrict__ part_em,
                     const float* __restrict__ part_ortho,
                     float* __restrict__ out_link, float* __restrict__ out_ent,
                     float* __restrict__ out_ortho) {
  __shared__ float sh[256];
  __shared__ float ent_sh;
  int t = threadIdx.x;
  if (t == 0) ent_sh = 0.0f;
  __syncthreads();
  for (int b = 0; b < B_; b++) {
    float v = colsum[b * C_ + t];
    sh[t] = v; __syncthreads();
    for (int o = 128; o; o >>= 1) { if (t < o) sh[t] += sh[t + o]; __syncthreads(); }
    float tot = sh[0]; __syncthreads();
    float cp = v / (tot + 1e-8f);
    sh[t] = -cp * logf(cp + 1e-8f); __syncthreads();
    for (int o = 128; o; o >>= 1) { if (t < o) sh[t] += sh[t + o]; __syncthreads(); }
    if (t == 0) ent_sh += sh[0];
    __syncthreads();
  }
  if (t == 0) {
    *out_ent = ent_sh / (float)B_;
    float link = 0.0f, orto = 0.0f;
    for (int b = 0; b < B_; b++) {
      float emv = part_em[b];
      link += (emv > 0.0f) ? part_nll[b] / fmaxf(emv, 1.0f) : 0.0f;
      orto += part_ortho[b] / (float)(C_ * C_);
    }
    *out_link = link / (float)B_;
    *out_ortho = orto / (float)B_;
  }
}

// ---------------------------------------------------------------------------
// Host orchestration (graph-capture safe: only kernel launches on `stream`)
// ---------------------------------------------------------------------------
extern "C" void kernel_launch(void* const* d_in, const int* in_sizes, int n_in,
                              void* d_out, int out_size, void* d_ws,
                              size_t ws_size, hipStream_t stream) {
  (void)in_sizes; (void)n_in; (void)out_size; (void)ws_size;
  const float* nf    = (const float*)d_in[0];
  const int*   ei    = (const int*)d_in[1];
  const float* nmask = (const float*)d_in[2];
  const float* emask = (const float*)d_in[3];
  const float* e0Ws  = (const float*)d_in[4];
  const float* e0Wn  = (const float*)d_in[5];
  const float* e0b   = (const float*)d_in[6];
  const float* e1Ws  = (const float*)d_in[7];
  const float* e1Wn  = (const float*)d_in[8];
  const float* e1b   = (const float*)d_in[9];
  const float* a0Ws  = (const float*)d_in[10];
  const float* a0Wn  = (const float*)d_in[11];
  const float* a0b   = (const float*)d_in[12];
  const float* a1Ws  = (const float*)d_in[13];
  const float* a1Wn  = (const float*)d_in[14];
  const float* a1b   = (const float*)d_in[15];
  const float* lpW1  = (const float*)d_in[16];
  const float* lpb1  = (const float*)d_in[17];
  const float* lpW2  = (const float*)d_in[18];
  const float* lpb2  = (const float*)d_in[19];
  float* out = (float*)d_out;
  float* ws  = (float*)d_ws;

  // workspace layout (floats); big buffers reuse across phases
  float* deg    = ws;                     // B*N
  float* colsum = ws + 16384;             // B*C
  float* part   = ws + 16384 + 2048;      // [0..7]=nll [8..15]=em [16..23]=ortho
  float* big    = ws + 32768;
  const size_t BIG = (size_t)B_ * N_ * 256;  // 4194304 floats
  float* agg   = big;                 // later reused as dense A (N*N == BIG)
  float* meanb = big + BIG;           // later reused as T (C*N)
  float* hbuf  = big + 2 * BIG;
  float* embb  = big + 3 * BIG;
  float* afb   = big + 4 * BIG;       // later: padj (B*C*C) + gram (B*C*C)
  float* alog  = big + 5 * BIG;
  float* Sb    = big + 6 * BIG;
  float* Abuf  = agg;
  float* Tbuf  = meanb;
  float* padj  = afb;
  float* gram  = afb + (size_t)B_ * C_ * C_;

  auto fill = [&](float* p, float v, size_t n) {
    size_t bl = (n + 255) / 256;
    int blocks = (int)(bl < 4096 ? bl : 4096);
    fill_f32<<<blocks, 256, 0, stream>>>(p, v, n);
  };

  fill(ws, 0.0f, 32768);  // deg + colsum + partials
  deg_kernel<<<(B_ * E_ + 255) / 256, 256, 0, stream>>>(ei, emask, deg);

  auto sage = [&](const float* X, int K, const float* Wsm, const float* Wnm,
                  const float* bias, float* o) {
    fill(agg, 0.0f, (size_t)B_ * N_ * K);
    size_t tot = (size_t)B_ * E_ * (K / 4);
    aggregate_kernel<<<(int)((tot + 255) / 256), 256, 0, stream>>>(X, ei, emask, agg, K);
    size_t tot2 = (size_t)B_ * N_ * K;
    mean_div_kernel<<<(int)((tot2 + 255) / 256), 256, 0, stream>>>(agg, deg, meanb, K);
    sage_gemm_wmma<<<dim3(N_ / 16, F_ / 64, B_), 32, 0, stream>>>(X, meanb, Wsm,
                                                                  Wnm, bias,
                                                                  nmask, o, K);
  };
  sage(nf,   DIN, e0Ws, e0Wn, e0b, hbuf);
  sage(hbuf, 256, e1Ws, e1Wn, e1b, embb);
  sage(nf,   DIN, a0Ws, a0Wn, a0b, afb);
  sage(afb,  256, a1Ws, a1Wn, a1b, alog);

  softmax_kernel<<<B_ * N_, 256, 0, stream>>>(alog, nmask, Sb);
  colsum_kernel<<<(B_ * C_) / 256, 256, 0, stream>>>(Sb, colsum, out + OUT_PNM);

  // pooled_feats[b] = S_b^T @ emb_b  (C x Dout), straight into d_out
  gemm_tn_wmma<<<dim3(C_ / 16, F_ / 64, B_), 32, 0, stream>>>(
      Sb, embb, out + OUT_PF, N_, C_, F_, F_,
      (size_t)N_ * C_, (size_t)N_ * F_, (size_t)C_ * F_);
  // gram[b] = S_b^T @ S_b
  gemm_tn_wmma<<<dim3(C_ / 16, C_ / 64, B_), 32, 0, stream>>>(
      Sb, Sb, gram, N_, C_, C_, C_,
      (size_t)N_ * C_, (size_t)N_ * C_, (size_t)C_ * C_);
  reduce_ortho_kernel<<<dim3(C_ * C_ / 256, B_), 256, 0, stream>>>(gram, part + 16);

  // pooled_adj[b] = S_b^T @ A_b @ S_b, dense A rebuilt per batch (L2-resident)
  for (int b = 0; b < B_; b++) {
    fill(Abuf, 0.0f, (size_t)N_ * N_);
    scatter_adj_kernel<<<E_ / 256, 256, 0, stream>>>(ei, emask, Abuf, b);
    gemm_tn_wmma<<<dim3(C_ / 16, N_ / 64, 1), 32, 0, stream>>>(
        Sb + (size_t)b * N_ * C_, Abuf, Tbuf, N_, C_, N_, N_, 0, 0, 0);
    gemm_nn_wmma<<<dim3(C_ / 16, C_ / 64, 1), 32, 0, stream>>>(
        Tbuf, Sb + (size_t)b * N_ * C_, padj + (size_t)b * C_ * C_,
        N_, N_, C_, C_);
  }

  fill(out + OUT_PEI, 0.0f, (size_t)B_ * 2 * C_ * C_);
  fill(out + OUT_PEM, 0.0f, (size_t)B_ * C_ * C_);
  compact_kernel<<<B_, 1024, 0, stream>>>(padj, out + OUT_PEI, out + OUT_PEM);

  linkpred_kernel<<<dim3(E_ / 16, B_), 32, 0, stream>>>(
      embb, ei, emask, lpW1, lpb1, lpW2, lpb2, part);
  emsum_kernel<<<B_, 256, 0, stream>>>(emask, part + 8);
  finalize_kernel<<<1, 256, 0, stream>>>(colsum, part, part + 8, part + 16,
                                         out + OUT_LINK, out + OUT_ENT,
                                         out + OUT_ORTHO);
}